// GraphTransformer2Layer_37082747633987
// MI455X (gfx1250) — compile-verified
//
#include <hip/hip_runtime.h>
#include <hip/hip_bf16.h>
#include <math.h>

// ---------------------------------------------------------------------------
// GraphTransformer (2-layer TransformerConv) for MI455X / gfx1250 (wave32).
// Dense projections: V_WMMA_F32_16X16X4_F32 (full fp32 matrix pipe), K loop
// software-pipelined (double-buffered fragments) with 32-bit offset
// addressing.  Edge phase: 2 fused passes (logits+max, exp+sum+scatter);
// softmax division deferred to node-level finalize.
// ---------------------------------------------------------------------------

typedef __attribute__((ext_vector_type(2))) float v2f;
typedef __attribute__((ext_vector_type(8))) float v8f;

#define IN_DIM 128
#define D1     256
#define HEADS  8
#define HID    32
#define NCLS   10
#define LD2    16   // padded layer-2 channel stride

// ---------------------------------------------------------------------------
// fp32 WMMA GEMM: C[M x ldc] = A[M x Kin] * W[Kin x Nlog] + bias.
// One wave computes a 16 x (16*NT) tile.  PAD=false: all columns valid,
// zero guards.  PAD=true: columns >= Nlog are exact zeros via clamped
// index + 0/1 mask (branchless; EXEC stays all-1s for every WMMA).
// ---------------------------------------------------------------------------
template <int NT, bool PAD>
__global__ __launch_bounds__(256) void wmma_gemm_f32(
    const float* __restrict__ A, const float* __restrict__ W,
    const float* __restrict__ bias, float* __restrict__ C,
    int M, int Kin, int Nlog, int ldc)
{
    const int nGroups = ldc / (16 * NT);
    const int lane = threadIdx.x & 31;
    const int wave = blockIdx.x * (blockDim.x >> 5) + (threadIdx.x >> 5);
    const int mTiles = (M + 15) >> 4;
    const int mTile  = wave / nGroups;
    const int nGroup = wave - mTile * nGroups;
    if (mTile >= mTiles) return;                 // wave-uniform exit

    const int m0 = mTile << 4;
    const int n0 = nGroup * (NT * 16);

    const int rc   = lane & 15;                  // row (A) / col (B,C) in tile
    const int koff = (lane >> 4) << 1;           // K sub-offset per half-wave
    const int arow = min(m0 + rc, M - 1);

    // 32-bit offsets from uniform bases (all matrices < 2^31 elements):
    // encourages SADDR + voffset addressing, no per-step 64-bit chains.
    const int aoff = arow * Kin + koff;
    const int nbase = n0 + rc;
    int col; float mskv;
    if (PAD) { col = min(nbase, Nlog - 1); mskv = (nbase < Nlog) ? 1.f : 0.f; }
    else     { col = nbase;                mskv = 1.f; }
    int woff = koff * Nlog + col;
    const int wstep = 4 * Nlog;

    v8f acc[NT];
#pragma unroll
    for (int t = 0; t < NT; ++t)
        acc[t] = (v8f){0.f, 0.f, 0.f, 0.f, 0.f, 0.f, 0.f, 0.f};

    // ---- software-pipelined K loop (prefetch one k-step ahead) ----
    v2f a_cur, b_cur[NT];
    a_cur.x = A[aoff];
    a_cur.y = A[aoff + 1];
#pragma unroll
    for (int t = 0; t < NT; ++t) {
        b_cur[t].x = W[woff + t * 16];
        b_cur[t].y = W[woff + Nlog + t * 16];
        if (PAD) { b_cur[t].x *= mskv; b_cur[t].y *= mskv; }
    }

    for (int k0 = 4; k0 < Kin; k0 += 4) {
        woff += wstep;
        v2f a_nxt, b_nxt[NT];
        a_nxt.x = A[aoff + k0];
        a_nxt.y = A[aoff + k0 + 1];
#pragma unroll
        for (int t = 0; t < NT; ++t) {
            b_nxt[t].x = W[woff + t * 16];
            b_nxt[t].y = W[woff + Nlog + t * 16];
            if (PAD) { b_nxt[t].x *= mskv; b_nxt[t].y *= mskv; }
        }
#pragma unroll
        for (int t = 0; t < NT; ++t)
            acc[t] = __builtin_amdgcn_wmma_f32_16x16x4_f32(
                false, a_cur, false, b_cur[t], (short)0, acc[t], false, false);
        a_cur = a_nxt;
#pragma unroll
        for (int t = 0; t < NT; ++t) b_cur[t] = b_nxt[t];
    }
#pragma unroll
    for (int t = 0; t < NT; ++t)
        acc[t] = __builtin_amdgcn_wmma_f32_16x16x4_f32(
            false, a_cur, false, b_cur[t], (short)0, acc[t], false, false);

    // C/D layout: VGPR v, lanes 0-15 -> M=v, lanes 16-31 -> M=v+8, N=lane&15
    const int mo = (lane >> 4) << 3;
    if (m0 + 16 <= M) {                          // wave-uniform fast path
#pragma unroll
        for (int t = 0; t < NT; ++t) {
            const int n  = n0 + t * 16 + rc;
            const float bv = PAD ? ((n < Nlog) ? bias[n] : 0.f) : bias[n];
#pragma unroll
            for (int v = 0; v < 8; ++v)
                C[(size_t)(m0 + mo + v) * ldc + n] = acc[t][v] + bv;
        }
    } else {
#pragma unroll
        for (int t = 0; t < NT; ++t) {
            const int n  = n0 + t * 16 + rc;
            const float bv = (n < Nlog) ? bias[n] : 0.f;
#pragma unroll
            for (int v = 0; v < 8; ++v) {
                const int m = m0 + mo + v;
                if (m < M) C[(size_t)m * ldc + n] = acc[t][v] + bv;
            }
        }
    }
}

// ---------------------------------------------------------------------------
// Utility fill
// ---------------------------------------------------------------------------
__global__ void fill_kernel(float* __restrict__ p, float v, long n)
{
    long i = (long)blockIdx.x * blockDim.x + threadIdx.x;
    if (i < n) p[i] = v;
}

// ---------------------------------------------------------------------------
// Edge pass 1: logits + segment-max (fused).
// Per-head rows are 16-float aligned in both layers -> float4 loads are safe.
// ---------------------------------------------------------------------------
template <int H, int C>
__global__ void edge_logits_max_k(const int* __restrict__ src, const int* __restrict__ dst,
                                  const float* __restrict__ q, const float* __restrict__ k,
                                  float* __restrict__ logits, float* __restrict__ m,
                                  int E, int ld, float scale)
{
    long idx = (long)blockIdx.x * blockDim.x + threadIdx.x;
    if (idx >= (long)E * H) return;
    const int e = (int)(idx / H);
    const int h = (int)(idx - (long)e * H);
    const int s = src[e], d = dst[e];
    const float* qp = q + (size_t)d * ld + h * C;
    const float* kp = k + (size_t)s * ld + h * C;
    const float4* q4 = reinterpret_cast<const float4*>(qp);
    const float4* k4 = reinterpret_cast<const float4*>(kp);
    float acc = 0.f;
#pragma unroll
    for (int c = 0; c < C / 4; ++c) {
        const float4 a = q4[c], b = k4[c];
        acc = fmaf(a.x, b.x, acc); acc = fmaf(a.y, b.y, acc);
        acc = fmaf(a.z, b.z, acc); acc = fmaf(a.w, b.w, acc);
    }
#pragma unroll
    for (int c = (C / 4) * 4; c < C; ++c) acc = fmaf(qp[c], kp[c], acc);
    const float lg = acc * scale;
    logits[idx] = lg;
    atomicMax(&m[(size_t)d * H + h], lg);       // global_atomic_max_num_f32
}

// ---------------------------------------------------------------------------
// Edge pass 2: ex = exp(logit - m[dst]); sum[dst] += ex;
// agg[dst] += ex * v[src]  (unnormalized; division deferred to finalize).
// ---------------------------------------------------------------------------
template <int H, int C>
__global__ void edge_msg_k(const int* __restrict__ src, const int* __restrict__ dst,
                           const float* __restrict__ logits, const float* __restrict__ m,
                           float* __restrict__ sum, const float* __restrict__ v,
                           float* __restrict__ agg, int E, int ld)
{
    long idx = (long)blockIdx.x * blockDim.x + threadIdx.x;
    if (idx >= (long)E * H) return;
    const int e = (int)(idx / H);
    const int h = (int)(idx - (long)e * H);
    const int s = src[e], d = dst[e];
    const float ex = __expf(logits[idx] - m[(size_t)d * H + h]);
    atomicAdd(&sum[(size_t)d * H + h], ex);
    const float* vp = v   + (size_t)s * ld + h * C;
    float*       op = agg + (size_t)d * ld + h * C;
    const float4* v4 = reinterpret_cast<const float4*>(vp);
#pragma unroll
    for (int c = 0; c < C / 4; ++c) {
        const float4 vv = v4[c];
        atomicAdd(&op[c * 4 + 0], ex * vv.x);   // global_atomic_add_f32
        atomicAdd(&op[c * 4 + 1], ex * vv.y);
        atomicAdd(&op[c * 4 + 2], ex * vv.z);
        atomicAdd(&op[c * 4 + 3], ex * vv.w);
    }
#pragma unroll
    for (int c = (C / 4) * 4; c < C; ++c)
        atomicAdd(&op[c], ex * vp[c]);
}

// ---------------------------------------------------------------------------
// Layer-1 finalize: h = elu(skip + agg/(sum+eps)), one thread per (node,head).
// ---------------------------------------------------------------------------
template <int H, int C>
__global__ void finalize_elu_k(float* __restrict__ h, const float* __restrict__ agg,
                               const float* __restrict__ sum, int N, int ld)
{
    long idx = (long)blockIdx.x * blockDim.x + threadIdx.x;
    if (idx >= (long)N * H) return;
    const int n  = (int)(idx / H);
    const int hh = (int)(idx - (long)n * H);
    const float inv = 1.f / (sum[idx] + 1e-16f);
    float*       hp = h   + (size_t)n * ld + hh * C;
    const float* ap = agg + (size_t)n * ld + hh * C;
#pragma unroll
    for (int c = 0; c < C; ++c) {
        const float val = fmaf(ap[c], inv, hp[c]);
        hp[c] = (val > 0.f) ? val : expm1f(val);
    }
}

// ---------------------------------------------------------------------------
// Layer-2 finalize: out[n,c] = skip[n,c] + agg[n,c]/(sum[n]+eps), c < 10,
// compacting the padded 16-stride buffers directly into d_out.
// ---------------------------------------------------------------------------
__global__ void finalize_out_k(const float* __restrict__ skip, const float* __restrict__ agg,
                               const float* __restrict__ sum, float* __restrict__ out, int N)
{
    long i = (long)blockIdx.x * blockDim.x + threadIdx.x;
    if (i >= (long)N * NCLS) return;
    const int n = (int)(i / NCLS);
    const int c = (int)(i - (long)n * NCLS);
    const float inv = 1.f / (sum[n] + 1e-16f);
    out[i] = fmaf(agg[(size_t)n * LD2 + c], inv, skip[(size_t)n * LD2 + c]);
}

// ---------------------------------------------------------------------------
// Launcher
// ---------------------------------------------------------------------------
extern "C" void kernel_launch(void* const* d_in, const int* in_sizes, int n_in,
                              void* d_out, int out_size, void* d_ws, size_t ws_size,
                              hipStream_t stream)
{
    const float* x   = (const float*)d_in[0];
    const int*   ei  = (const int*)  d_in[1];
    const float* w1q = (const float*)d_in[2];  const float* b1q = (const float*)d_in[3];
    const float* w1k = (const float*)d_in[4];  const float* b1k = (const float*)d_in[5];
    const float* w1v = (const float*)d_in[6];  const float* b1v = (const float*)d_in[7];
    const float* w1s = (const float*)d_in[8];  const float* b1s = (const float*)d_in[9];
    const float* w2q = (const float*)d_in[10]; const float* b2q = (const float*)d_in[11];
    const float* w2k = (const float*)d_in[12]; const float* b2k = (const float*)d_in[13];
    const float* w2v = (const float*)d_in[14]; const float* b2v = (const float*)d_in[15];
    const float* w2s = (const float*)d_in[16]; const float* b2s = (const float*)d_in[17];
    float* out = (float*)d_out;

    const int N = in_sizes[0] / IN_DIM;
    const int E = in_sizes[1] / 2;
    const int* src = ei;       // edge_index[0]
    const int* dst = ei + E;   // edge_index[1]

    // ---- workspace layout (floats) ----
    float* p  = (float*)d_ws;
    float* q1   = p;           p += (size_t)N * D1;
    float* k1   = p;           p += (size_t)N * D1;
    float* v1   = p;           p += (size_t)N * D1;
    float* h1   = p;           p += (size_t)N * D1;  // skip, then finalized h1
    float* agg1 = p;           p += (size_t)N * D1;  // zeroed message accum
    float* alpha1 = p;         p += (size_t)E * HEADS;
    float* m1   = p;           p += (size_t)N * HEADS;
    float* sum1 = p;           /* end of live layer-1 state */
    // layer-2 buffers reuse q1 region (dead after layer-1 edge pass 2)
    float* q2   = q1;
    float* k2   = q2 + (size_t)N * LD2;
    float* v2   = k2 + (size_t)N * LD2;
    float* s2   = v2 + (size_t)N * LD2;              // skip
    float* agg2 = s2 + (size_t)N * LD2;              // zeroed message accum
    float* alpha2 = agg2 + (size_t)N * LD2;
    float* m2   = alpha2 + (size_t)E;
    float* sum2 = m2 + (size_t)N;

    const int B = 256;
    const float scale1 = 0.17677669529663687f;   // 1/sqrt(32)
    const float scale2 = 0.31622776601683794f;   // 1/sqrt(10)

    // ---- layer 1 projections (fp32 WMMA, no guards) ----
    {
        const int waves = ((N + 15) / 16) * (D1 / 64);
        const int grid  = (waves + 7) / 8;
        wmma_gemm_f32<4, false><<<grid, B, 0, stream>>>(x, w1q, b1q, q1, N, IN_DIM, D1, D1);
        wmma_gemm_f32<4, false><<<grid, B, 0, stream>>>(x, w1k, b1k, k1, N, IN_DIM, D1, D1);
        wmma_gemm_f32<4, false><<<grid, B, 0, stream>>>(x, w1v, b1v, v1, N, IN_DIM, D1, D1);
        wmma_gemm_f32<4, false><<<grid, B, 0, stream>>>(x, w1s, b1s, h1, N, IN_DIM, D1, D1);
    }

    // ---- layer 1 attention (2 edge passes + node finalize) ----
    {
        const long NH = (long)N * HEADS;
        const long EH = (long)E * HEADS;
        const long ND = (long)N * D1;
        fill_kernel<<<(int)((NH + B - 1) / B), B, 0, stream>>>(m1, -INFINITY, NH);
        fill_kernel<<<(int)((NH + B - 1) / B), B, 0, stream>>>(sum1, 0.f, NH);
        fill_kernel<<<(int)((ND + B - 1) / B), B, 0, stream>>>(agg1, 0.f, ND);
        edge_logits_max_k<HEADS, HID><<<(int)((EH + B - 1) / B), B, 0, stream>>>(
            src, dst, q1, k1, alpha1, m1, E, D1, scale1);
        edge_msg_k<HEADS, HID><<<(int)((EH + B - 1) / B), B, 0, stream>>>(
            src, dst, alpha1, m1, sum1, v1, agg1, E, D1);
        finalize_elu_k<HEADS, HID><<<(int)((NH + B - 1) / B), B, 0, stream>>>(
            h1, agg1, sum1, N, D1);
    }

    // ---- layer 2 projections (fp32 WMMA, 10 cols padded to 16, branchless) ----
    {
        const int waves = (N + 15) / 16;       // nGroups == 1
        const int grid  = (waves + 7) / 8;
        wmma_gemm_f32<1, true><<<grid, B, 0, stream>>>(h1, w2q, b2q, q2, N, D1, NCLS, LD2);
        wmma_gemm_f32<1, true><<<grid, B, 0, stream>>>(h1, w2k, b2k, k2, N, D1, NCLS, LD2);
        wmma_gemm_f32<1, true><<<grid, B, 0, stream>>>(h1, w2v, b2v, v2, N, D1, NCLS, LD2);
        wmma_gemm_f32<1, true><<<grid, B, 0, stream>>>(h1, w2s, b2s, s2, N, D1, NCLS, LD2);
    }

    // ---- layer 2 attention (H=1) ----
    {
        const long NP = (long)N * LD2;
        fill_kernel<<<(N + B - 1) / B, B, 0, stream>>>(m2, -INFINITY, (long)N);
        fill_kernel<<<(N + B - 1) / B, B, 0, stream>>>(sum2, 0.f, (long)N);
        fill_kernel<<<(int)((NP + B - 1) / B), B, 0, stream>>>(agg2, 0.f, NP);
        edge_logits_max_k<1, NCLS><<<(E + B - 1) / B, B, 0, stream>>>(
            src, dst, q2, k2, alpha2, m2, E, LD2, scale2);
        edge_msg_k<1, NCLS><<<(E + B - 1) / B, B, 0, stream>>>(
            src, dst, alpha2, m2, sum2, v2, agg2, E, LD2);
        const long NO = (long)N * NCLS;
        finalize_out_k<<<(int)((NO + B - 1) / B), B, 0, stream>>>(s2, agg2, sum2, out, N);
    }
}